// RNNClassifier_84799834292822
// MI455X (gfx1250) — compile-verified
//
#include <hip/hip_runtime.h>
#include <cstddef>

typedef __attribute__((ext_vector_type(2))) float v2f;
typedef __attribute__((ext_vector_type(8))) float v8f;

#define WPB      2     // waves per block in the GEMM kernel
#define A_STRIDE 258   // padded LDS row stride (floats); 258 mod 64 = 2 -> ~2-way worst case

__device__ __forceinline__ float sigmoidf(float x) {
    return 1.0f / (1.0f + __expf(-x));
}

// out[r][n] = sum_{k<256} sigmoid(emb[tokens[r]][k]) * W[k][n]   (n < 4)
// One wave per 16-row tile; K=256 via 64 x V_WMMA_F32_16X16X4_F32 (full f32 precision).
__global__ void __launch_bounds__(WPB * 32)
embed_sigmoid_gemm_kernel(const int* __restrict__ tokens,   // [R] token ids (int32)
                          const float* __restrict__ emb,    // [32000, 256]
                          const float* __restrict__ W,      // [260, 4]; rows 0..255 used
                          float* __restrict__ out,          // [R, 4]
                          int ntiles)
{
    __shared__ float aLds[WPB * 16 * A_STRIDE];   // sigmoid'd A tiles, one per wave
    __shared__ v2f   bLds[64 * 2 * 16];           // padded B fragments, shared (16 KB)

    const int lane = threadIdx.x & 31;
    const int wave = threadIdx.x >> 5;
    const int tile = blockIdx.x * WPB + wave;
    const bool live = (tile < ntiles);            // grids are exact; stays true
    const int tileC = live ? tile : (ntiles - 1); // clamp so staging never goes OOB
    const int rowBase = tileC * 16;

    // ---- Stage padded B fragments once per block (no branches in hot loop) ----
    // bLds[(kk*2 + h)*16 + n] = { W[kk*4+2h][n], W[kk*4+2h+1][n] },  zero for n >= 4.
    for (int i = threadIdx.x; i < 64 * 2 * 16; i += WPB * 32) {
        const int n  = i & 15;
        const int k0 = (i >> 4) * 2;              // 0,2,4,...,254
        v2f v = {0.f, 0.f};
        if (n < 4) {
            v.x = W[k0 * 4 + n];
            v.y = W[(k0 + 1) * 4 + n];
        }
        bLds[i] = v;
    }

    // ---- Stage 16 sigmoid'd embedding rows into LDS (1KB coalesced per row) ----
    float* tl = aLds + wave * (16 * A_STRIDE);
    for (int m = 0; m < 16; ++m) {
        const int tok = tokens[rowBase + m];
        const float4* src = (const float4*)(emb + (size_t)tok * 256) + lane * 2;
        float4 v0 = src[0];
        float4 v1 = src[1];
        v0.x = sigmoidf(v0.x); v0.y = sigmoidf(v0.y); v0.z = sigmoidf(v0.z); v0.w = sigmoidf(v0.w);
        v1.x = sigmoidf(v1.x); v1.y = sigmoidf(v1.y); v1.z = sigmoidf(v1.z); v1.w = sigmoidf(v1.w);
        float* dst = tl + m * A_STRIDE + lane * 8;
        *(float4*)dst       = v0;
        *(float4*)(dst + 4) = v1;
    }
    __syncthreads();

    // f32 WMMA 16x16x4 fragment mapping (ISA 7.12.2):
    //   A 16x4: lane L holds row M=L&15, K = khalf+{0,1} (khalf = 2*(L>>4))
    //   B 4x16: lane L holds col N=L&15, same K split
    const int n     = lane & 15;
    const int half  = lane >> 4;                  // 0 or 1
    const float* ap = tl + n * A_STRIDE + half * 2;   // note: row index == lane&15
    const v2f* bp   = bLds + half * 16 + n;

    // Two accumulators (even/odd kk) break the WMMA->WMMA RAW hazard chain.
    v8f acc0 = {0.f, 0.f, 0.f, 0.f, 0.f, 0.f, 0.f, 0.f};
    v8f acc1 = {0.f, 0.f, 0.f, 0.f, 0.f, 0.f, 0.f, 0.f};
    #pragma unroll 8
    for (int kk = 0; kk < 64; kk += 2) {
        v2f a0 = *(const v2f*)(ap + kk * 4);          // ds_load_b64
        v2f b0 = bp[kk * 32];                          // ds_load_b64, conflict-free
        acc0 = __builtin_amdgcn_wmma_f32_16x16x4_f32(false, a0, false, b0, (short)0, acc0,
                                                     false, false);
        v2f a1 = *(const v2f*)(ap + kk * 4 + 4);
        v2f b1 = bp[kk * 32 + 32];
        acc1 = __builtin_amdgcn_wmma_f32_16x16x4_f32(false, a1, false, b1, (short)0, acc1,
                                                     false, false);
    }
    v8f acc = acc0 + acc1;

    // C layout: VGPR v, lanes 0-15 -> M=v, lanes 16-31 -> M=v+8; N = lane&15.
    if (live && n < 4) {
        const int mb = half * 8;
        #pragma unroll
        for (int v = 0; v < 8; ++v)
            out[(size_t)(rowBase + mb + v) * 4 + n] = acc[v];
    }
}

// Sequential part: h_{t+1}[b] = a[t,b] + bh + sigmoid(h_t[b]) @ Whh (4x4),
// then fused classifier head. 256 independent chains, one per thread.
__global__ void __launch_bounds__(32)
rnn_scan_kernel(const float* __restrict__ a_ws,   // [2048, 256, 4]
                const float* __restrict__ axo,    // [256, 4]  (x-part of final output step)
                const float* __restrict__ h0,     // [256, 4]
                const float* __restrict__ Wh,     // [260, 4]
                const float* __restrict__ bh,     // [4]
                const float* __restrict__ Wo,     // [260, 4]
                const float* __restrict__ bo,     // [4]
                const float* __restrict__ Wfc,    // [4, 10]
                const float* __restrict__ bfc,    // [10]
                float* __restrict__ logits)       // [256, 10]
{
    const int b = blockIdx.x * 32 + threadIdx.x;  // 8 blocks x 32 = 256

    float h[4], bhv[4], Whh[4][4];
    #pragma unroll
    for (int j = 0; j < 4; ++j) {
        h[j]   = h0[b * 4 + j];
        bhv[j] = bh[j];
        #pragma unroll
        for (int i = 0; i < 4; ++i)
            Whh[i][j] = Wh[(256 + i) * 4 + j];
    }

    // Steps t = 0..2046 update the carry; step 2047 only produces the output.
    for (int t = 0; t < 2047; ++t) {
        const float4 a4 = *(const float4*)(a_ws + ((size_t)t * 256 + b) * 4);
        const float s0 = sigmoidf(h[0]);
        const float s1 = sigmoidf(h[1]);
        const float s2 = sigmoidf(h[2]);
        const float s3 = sigmoidf(h[3]);
        h[0] = a4.x + bhv[0] + s0 * Whh[0][0] + s1 * Whh[1][0] + s2 * Whh[2][0] + s3 * Whh[3][0];
        h[1] = a4.y + bhv[1] + s0 * Whh[0][1] + s1 * Whh[1][1] + s2 * Whh[2][1] + s3 * Whh[3][1];
        h[2] = a4.z + bhv[2] + s0 * Whh[0][2] + s1 * Whh[1][2] + s2 * Whh[2][2] + s3 * Whh[3][2];
        h[3] = a4.w + bhv[3] + s0 * Whh[0][3] + s1 * Whh[1][3] + s2 * Whh[2][3] + s3 * Whh[3][3];
    }

    // Final step output: o = axo + bo + sigmoid(h_2047) @ Wo[256:260,:]
    float s[4], o[4];
    #pragma unroll
    for (int i = 0; i < 4; ++i) s[i] = sigmoidf(h[i]);
    #pragma unroll
    for (int j = 0; j < 4; ++j) {
        float v = axo[b * 4 + j] + bo[j];
        #pragma unroll
        for (int i = 0; i < 4; ++i)
            v += s[i] * Wo[(256 + i) * 4 + j];
        o[j] = v;
    }

    // logits = o @ Wfc + bfc
    #pragma unroll
    for (int c = 0; c < 10; ++c) {
        float acc = bfc[c];
        #pragma unroll
        for (int j = 0; j < 4; ++j)
            acc += o[j] * Wfc[j * 10 + c];
        logits[b * 10 + c] = acc;
    }
}

extern "C" void kernel_launch(void* const* d_in, const int* in_sizes, int n_in,
                              void* d_out, int out_size, void* d_ws, size_t ws_size,
                              hipStream_t stream) {
    (void)in_sizes; (void)n_in; (void)out_size; (void)ws_size;

    const int*   x   = (const int*)d_in[0];   // [2048, 256] token ids
    // d_in[1] = original_lens (unused by the reference)
    const float* emb = (const float*)d_in[2]; // [32000, 256]
    const float* h0  = (const float*)d_in[3]; // [256, 4]
    const float* Wh  = (const float*)d_in[4]; // [260, 4]
    const float* bh  = (const float*)d_in[5]; // [4]
    const float* Wo  = (const float*)d_in[6]; // [260, 4]
    const float* bo  = (const float*)d_in[7]; // [4]
    const float* Wfc = (const float*)d_in[8]; // [4, 10]
    const float* bfc = (const float*)d_in[9]; // [10]
    float* out = (float*)d_out;               // [256, 10]

    float* a_ws = (float*)d_ws;                       // 2048*256*4 f32 = 8 MB
    float* axo  = a_ws + (size_t)2048 * 256 * 4;      // 256*4 f32

    // Pass 1: a[t,b,:] = sigmoid(emb[x[t,b]]) @ Wh[0:256,:]   (all 524288 rows)
    {
        const int ntiles = (2048 * 256) / 16;         // 32768, multiple of WPB
        embed_sigmoid_gemm_kernel<<<ntiles / WPB, WPB * 32, 0, stream>>>(
            x, emb, Wh, a_ws, ntiles);
    }
    // Pass 2: axo[b,:] = sigmoid(emb[x[2047,b]]) @ Wo[0:256,:]  (last timestep only)
    {
        const int ntiles = 256 / 16;                  // 16, multiple of WPB
        embed_sigmoid_gemm_kernel<<<ntiles / WPB, WPB * 32, 0, stream>>>(
            x + 2047 * 256, emb, Wo, axo, ntiles);
    }
    // Pass 3: sequential 4-wide recurrence + classifier head
    rnn_scan_kernel<<<8, 32, 0, stream>>>(a_ws, axo, h0, Wh, bh, Wo, bo, Wfc, bfc, out);
}